// VectorQuantizer_41068477284657
// MI455X (gfx1250) — compile-verified
//
#include <hip/hip_runtime.h>
#include <hip/hip_bf16.h>

// ---------------------------------------------------------------------------
// VQ-VAE vector quantizer for MI455X (gfx1250, wave32, WMMA).
//
// dist argmin over codes n:  argmin_n ( ||e_n||^2 - 2 * z.e_n )   (||z||^2 is
// row-constant and dropped).  The 16384x8192 (K=64) dot-product GEMM is done
// with V_WMMA_F32_16X16X4_F32 (full f32 path, layouts per ISA 7.12.2), fused
// with a running per-row argmin so the 536MB distance matrix is never stored.
// ---------------------------------------------------------------------------

typedef __attribute__((ext_vector_type(2))) float v2f;
typedef __attribute__((ext_vector_type(8))) float v8f;

#define N_ROWS   16384     // B*H*W = 16*32*32
#define N_CODES  8192      // V
#define K_DIM    64        // C
#define HW       1024      // H*W
#define BC_STRIDE 65536    // C*H*W (per-batch stride in z / zq)
#define ZQ_ELEMS 1048576   // B*C*H*W
#define IDX_OFF  1048576
#define LOSS_OFF 1064960   // ZQ_ELEMS + N_ROWS

// -------------------------- codebook norms ---------------------------------
__global__ void vq_e2_kernel(const float* __restrict__ emb,
                             float* __restrict__ e2) {
  int v = blockIdx.x * blockDim.x + threadIdx.x;
  if (v >= N_CODES) return;
  const float4* p = (const float4*)(emb + v * K_DIM);
  float s = 0.0f;
#pragma unroll
  for (int i = 0; i < K_DIM / 4; ++i) {
    float4 q = p[i];
    s = fmaf(q.x, q.x, s); s = fmaf(q.y, q.y, s);
    s = fmaf(q.z, q.z, s); s = fmaf(q.w, q.w, s);
  }
  e2[v] = s;
}

// Scatter one quantized row (64 channels) to [B,C,H,W] layout, return the
// SSE contribution of the 2 channels this lane handles.
__device__ __forceinline__ float emit_row(int row, int code, int lane,
                                          const float* __restrict__ z,
                                          const float* __restrict__ emb,
                                          float* __restrict__ zq) {
  const int b = row >> 10;          // row / HW
  const int hw = row & (HW - 1);
  const float* e = emb + code * K_DIM;
  float q0 = e[lane];
  float q1 = e[lane + 32];
  const long o0 = (long)b * BC_STRIDE + (long)lane * HW + hw;
  const long o1 = o0 + 32l * HW;
  float z0 = z[o0];
  float z1 = z[o1];
  zq[o0] = q0;
  zq[o1] = q1;
  float d0 = q0 - z0, d1 = q1 - z1;
  return fmaf(d0, d0, d1 * d1);
}

// -------------------------- main fused kernel ------------------------------
// 4 waves / block, 1 wave == one 16-row M-tile.  grid = 256 blocks.
__global__ __launch_bounds__(128) void vq_main_kernel(
    const float* __restrict__ z, const float* __restrict__ emb,
    const float* __restrict__ e2, float* __restrict__ out,
    float* __restrict__ partial) {
  const int lane = threadIdx.x & 31;
  const int wave = threadIdx.x >> 5;
  const int tile = blockIdx.x * 4 + wave;
  const int r16  = lane & 15;       // N-column (B tile) / M-row (A tile)
  const int half = lane >> 4;       // selects K sub-pair per ISA layout
  const int rowbase = tile * 16;

  // ---- A tile: z rows [rowbase, rowbase+16) kept in registers.
  // A lane layout (ISA 7.12.2, 32-bit A 16x4): lane l holds row l&15,
  // VGPR0 = K = 4*step + 2*half, VGPR1 = K+1.
  float ar0[16], ar1[16];
  {
    const int row = rowbase + r16;
    const int b = row >> 10, hw = row & (HW - 1);
    const float* zp = z + (long)b * BC_STRIDE + hw;   // stride HW per channel
#pragma unroll
    for (int kk = 0; kk < 16; ++kk) {
      const int k = kk * 4 + 2 * half;
      ar0[kk] = zp[(long)k * HW];
      ar1[kk] = zp[(long)(k + 1) * HW];
    }
  }

  float best[8];
  int   bidx[8];
#pragma unroll
  for (int r = 0; r < 8; ++r) { best[r] = 3.4e38f; bidx[r] = 0; }

  // ---- stream the codebook: 512 chunks of 16 codes, K=64 via 16 WMMAs.
  for (int n0 = 0; n0 < N_CODES; n0 += 16) {
    const float e2v = e2[n0 + r16];                 // ||e_n||^2, n = col r16
    // B lane layout mirrors A: lane holds column n0+r16 of E^T, i.e. row
    // n0+r16 of emb, K-pair {4*step + 2*half, +1}  -> contiguous float2.
    const float2* bp = (const float2*)(emb + (long)(n0 + r16) * K_DIM + 2 * half);
    v8f acc = {};
#pragma unroll
    for (int kk = 0; kk < 16; ++kk) {
      float2 bv = bp[kk * 2];
      v2f a; a[0] = ar0[kk]; a[1] = ar1[kk];
      v2f bb; bb[0] = bv.x;  bb[1] = bv.y;
      acc = __builtin_amdgcn_wmma_f32_16x16x4_f32(
          /*neg_a=*/false, a, /*neg_b=*/false, bb,
          /*c_mod=*/(short)0, acc, /*reuse_a=*/false, /*reuse_b=*/false);
    }
    // D layout: acc[r] at lane l = dot[row r + 8*half][col r16]
    const int n = n0 + r16;
#pragma unroll
    for (int r = 0; r < 8; ++r) {
      float d = fmaf(-2.0f, acc[r], e2v);
      if (d < best[r]) { best[r] = d; bidx[r] = n; }   // strict < keeps lowest n
    }
  }

  // ---- argmin across the 16 lanes of each half (ties -> lower index,
  // matching jnp.argmin first-occurrence semantics).
#pragma unroll
  for (int r = 0; r < 8; ++r) {
    float v = best[r]; int i = bidx[r];
#pragma unroll
    for (int m = 8; m >= 1; m >>= 1) {
      float v2 = __shfl_xor(v, m, 32);
      int   i2 = __shfl_xor(i, m, 32);
      if (v2 < v || (v2 == v && i2 < i)) { v = v2; i = i2; }
    }
    best[r] = v; bidx[r] = i;
  }
  // Now all lanes of half h hold argmin for rows {r + 8h}.

  // ---- write idx (as float, concatenated-output convention).
#pragma unroll
  for (int r = 0; r < 8; ++r) {
    if (r16 == r)
      out[IDX_OFF + rowbase + r + 8 * half] = (float)bidx[r];
  }

  // ---- gather z_q = emb[idx], scatter to [B,C,H,W], accumulate SSE.
  float sse = 0.0f;
#pragma unroll
  for (int r = 0; r < 8; ++r) {
    const int codeLo = __shfl(bidx[r], 0, 32);    // row r      (half 0 result)
    const int codeHi = __shfl(bidx[r], 16, 32);   // row r + 8  (half 1 result)
    sse += emit_row(rowbase + r,     codeLo, lane, z, emb, out);
    sse += emit_row(rowbase + 8 + r, codeHi, lane, z, emb, out);
  }
#pragma unroll
  for (int m = 16; m >= 1; m >>= 1) sse += __shfl_xor(sse, m, 32);
  if (lane == 0) partial[tile] = sse;
}

// -------------------------- loss reduction ---------------------------------
__global__ void vq_loss_kernel(const float* __restrict__ partial,
                               float* __restrict__ out) {
  __shared__ float sm[256];
  float s = 0.0f;
  for (int i = threadIdx.x; i < 1024; i += 256) s += partial[i];
  sm[threadIdx.x] = s;
  __syncthreads();
  for (int st = 128; st > 0; st >>= 1) {
    if (threadIdx.x < st) sm[threadIdx.x] += sm[threadIdx.x + st];
    __syncthreads();
  }
  // vq_loss = codebook + 0.25*commit = 1.25 * mean((z_q - z)^2)
  if (threadIdx.x == 0) out[LOSS_OFF] = 1.25f * sm[0] / (float)ZQ_ELEMS;
}

// ---------------------------------------------------------------------------
extern "C" void kernel_launch(void* const* d_in, const int* in_sizes, int n_in,
                              void* d_out, int out_size, void* d_ws, size_t ws_size,
                              hipStream_t stream) {
  const float* z   = (const float*)d_in[0];   // [16,64,32,32]
  const float* emb = (const float*)d_in[1];   // [8192,64]
  float* out = (float*)d_out;                 // zq(1048576) | idx(16384) | loss(1)
  float* e2      = (float*)d_ws;              // 8192 floats
  float* partial = e2 + N_CODES;              // 1024 floats

  vq_e2_kernel<<<N_CODES / 256, 256, 0, stream>>>(emb, e2);
  vq_main_kernel<<<256, 128, 0, stream>>>(z, emb, e2, out, partial);
  vq_loss_kernel<<<1, 256, 0, stream>>>(partial, out);
}